// KLDLossGroup_18769007083971
// MI455X (gfx1250) — compile-verified
//
#include <hip/hip_runtime.h>
#include <hip/hip_bf16.h>
#include <math.h>

// Problem constants (from the reference)
#define P_ 4
#define B_ 8
#define N_ 65536
#define G_ 4
#define C_ 12
#define PROTO_ROWS 120   // C * PROTO_PER_CLASS

typedef float v2f __attribute__((ext_vector_type(2)));
typedef float v8f __attribute__((ext_vector_type(8)));

// Workspace layout (floats):
//   [0 .. 383]    lse[c][b][g]            (C*B*G = 384)
//   [384 .. 479]  counts[c][b] as int32   (C*B = 96)
//   [512 .. 2047] T[c][b][a][g]           (C*B*16 = 1536)
#define WS_LSE   0
#define WS_CNT   384
#define WS_T     512

__device__ __forceinline__ int proj_of_class(const float* __restrict__ gci, int c) {
    // proj_idx[c] = argmax_r gci[r, c] // G   over r in [0, P*G)
    int best_r = 0;
    float best = gci[c];
    for (int r = 1; r < P_ * G_; ++r) {
        float v = gci[r * C_ + c];
        if (v > best) { best = v; best_r = r; }
    }
    return best_r / G_;
}

// ---------------------------------------------------------------------------
// Pass 1: per (c,b): masked online logsumexp per group + pixel count.
// Also zeroes the T accumulators for pass 2.
// ---------------------------------------------------------------------------
__global__ void __launch_bounds__(256)
lse_count_kernel(const float* __restrict__ act,
                 const int*   __restrict__ labels,
                 const float* __restrict__ gci,
                 float* __restrict__ ws) {
    const int cb = blockIdx.x;            // 0..C*B-1
    const int c  = cb / B_;
    const int b  = cb % B_;
    const int proj = proj_of_class(gci, c);

    const float* x  = act + ((size_t)proj * B_ + b) * (size_t)N_ * G_;
    const int* lab  = labels + (size_t)b * N_;

    float m[G_], s[G_];
    #pragma unroll
    for (int g = 0; g < G_; ++g) { m[g] = -INFINITY; s[g] = 0.0f; }
    int cnt = 0;

    for (int n = threadIdx.x; n < N_; n += blockDim.x) {
        if (lab[n] == c + 1) {
            ++cnt;
            #pragma unroll
            for (int g = 0; g < G_; ++g) {
                float v = x[(size_t)n * G_ + g];
                if (v > m[g]) { s[g] = s[g] * __expf(m[g] - v) + 1.0f; m[g] = v; }
                else          { s[g] += __expf(v - m[g]); }
            }
        }
    }

    __shared__ float sm[256][G_];
    __shared__ float ss[256][G_];
    __shared__ int   sc[256];
    const int t = threadIdx.x;
    #pragma unroll
    for (int g = 0; g < G_; ++g) { sm[t][g] = m[g]; ss[t][g] = s[g]; }
    sc[t] = cnt;
    __syncthreads();

    for (int off = 128; off > 0; off >>= 1) {
        if (t < off) {
            sc[t] += sc[t + off];
            #pragma unroll
            for (int g = 0; g < G_; ++g) {
                float m1 = sm[t][g],       s1 = ss[t][g];
                float m2 = sm[t + off][g], s2 = ss[t + off][g];
                float mm = fmaxf(m1, m2);
                float sn = 0.0f;
                if (m1 > -INFINITY) sn += s1 * __expf(m1 - mm);
                if (m2 > -INFINITY) sn += s2 * __expf(m2 - mm);
                sm[t][g] = mm; ss[t][g] = sn;
            }
        }
        __syncthreads();
    }

    if (t == 0) {
        const int count = sc[0];
        ((int*)ws)[WS_CNT + cb] = count;
        #pragma unroll
        for (int g = 0; g < G_; ++g) {
            float l = (count > 0) ? (sm[0][g] + __logf(ss[0][g])) : 0.0f;
            ws[WS_LSE + cb * G_ + g] = l;
        }
        for (int i = 0; i < 16; ++i) ws[WS_T + cb * 16 + i] = 0.0f;
    }
}

// ---------------------------------------------------------------------------
// Pass 2: T[c,b,a,g] = sum_n p[n,a]*lp[n,g] via WMMA Gram accumulation.
//   M-dim = a (rows 0..3 live), N-dim = g (cols 0..3 live), K = pixels.
//   32-bit 16x4 A layout: lane L (col=L&15, half=L>>4), VGPR j holds K=2*half+j.
//   B 4x16 uses the mirrored layout -> identical per-lane addressing.
// Branchless masking (bitwise &, no short-circuit), 4x unroll (16 pixels /
// 4 WMMAs per iteration), two alternating accumulators to decouple the
// WMMA accumulation chains, prefetch of the activation stream.
// ---------------------------------------------------------------------------
__global__ void __launch_bounds__(256)
t_wmma_kernel(const float* __restrict__ act,
              const int*   __restrict__ labels,
              const float* __restrict__ gci,
              float* __restrict__ ws) {
    const int cb = blockIdx.x;
    const int c  = cb / B_;
    const int b  = cb % B_;
    const int proj = proj_of_class(gci, c);

    const float* x  = act + ((size_t)proj * B_ + b) * (size_t)N_ * G_;
    const int* lab  = labels + (size_t)b * N_;

    const int wave = threadIdx.x >> 5;
    const int lane = threadIdx.x & 31;
    const int col  = lane & 15;       // N-dim index (group g when < 4)
    const int kh   = lane >> 4;       // K half-select
    const int g    = col & 3;         // clamp for safe addressing
    const bool colv = (col < 4);
    const int  cp1  = c + 1;

    const float lse = ws[WS_LSE + cb * G_ + g];

    const int nwaves = blockDim.x >> 5;
    const int span   = N_ / nwaves;   // 8192 pixels per wave (mult of 16)
    const int base0  = wave * span;

    v8f acc0 = {0.f, 0.f, 0.f, 0.f, 0.f, 0.f, 0.f, 0.f};
    v8f acc1 = {0.f, 0.f, 0.f, 0.f, 0.f, 0.f, 0.f, 0.f};

    constexpr int UNROLL = 4;                 // 16 pixels per outer iteration
    for (int base = base0; base < base0 + span; base += 4 * UNROLL) {
        // Prefetch the activation stream four iteration-blocks (64 pixels) ahead.
        __builtin_prefetch(&x[(size_t)(base + 16 * UNROLL) * G_], 0, 1);

        v2f Av[UNROLL], Bv[UNROLL];
        #pragma unroll
        for (int u = 0; u < UNROLL; ++u) {
            #pragma unroll
            for (int j = 0; j < 2; ++j) {
                const int n  = base + 4 * u + 2 * kh + j;
                const float xv = x[(size_t)n * G_ + g];   // unconditional load
                const int   lv = lab[n];                  // unconditional load
                const bool live = colv & (lv == cp1);     // no short-circuit
                const float lp = xv - lse;
                const float p  = __expf(lp);
                Av[u][j] = live ? p  : 0.0f;              // v_cndmask
                Bv[u][j] = live ? lp : 0.0f;              // v_cndmask
            }
        }
        #pragma unroll
        for (int u = 0; u < UNROLL; u += 2) {
            acc0 = __builtin_amdgcn_wmma_f32_16x16x4_f32(
                false, Av[u],     false, Bv[u],     (short)0, acc0, false, false);
            acc1 = __builtin_amdgcn_wmma_f32_16x16x4_f32(
                false, Av[u + 1], false, Bv[u + 1], (short)0, acc1, false, false);
        }
    }

    acc0 += acc1;

    // D layout: lane L holds column col=L&15; VGPR r holds row M = r + 8*(L>>4).
    // Live tile is rows a=0..3 (acc0[0..3] on kh==0 lanes), cols g=0..3.
    if (lane < 4) {
        float* T = ws + WS_T + cb * 16;
        #pragma unroll
        for (int a = 0; a < 4; ++a)
            atomicAdd(&T[a * 4 + lane], acc0[a]);
    }
}

// ---------------------------------------------------------------------------
// Pass 3: symmetric KL per (c,b,j<k), exp(-kld), masked mean -> scalar.
// ---------------------------------------------------------------------------
__global__ void finalize_kernel(const float* __restrict__ proto,
                                const float* __restrict__ ws,
                                float* __restrict__ out) {
    __shared__ float stotal[C_ * B_];
    __shared__ int   sdenom[C_ * B_];
    const int t = threadIdx.x;         // launched with exactly C_*B_ threads
    float total = 0.0f;
    int   denom = 0;

    {
        const int c = t / B_;
        float colsum = 0.0f;
        for (int r = 0; r < PROTO_ROWS; ++r) colsum += proto[r * C_ + c];
        const bool proto_ok = colsum > 0.0f;
        const int count = ((const int*)ws)[WS_CNT + t];
        if (proto_ok && count >= 2) {
            const float* T = ws + WS_T + t * 16;
            const float diag[4] = { T[0], T[5], T[10], T[15] };
            #pragma unroll
            for (int j = 0; j < 4; ++j)
                #pragma unroll
                for (int k = j + 1; k < 4; ++k) {
                    float kld = 0.5f * (diag[j] + diag[k] - T[j * 4 + k] - T[k * 4 + j]);
                    total += __expf(-kld);
                    denom += 1;
                }
        }
    }
    stotal[t] = total;
    sdenom[t] = denom;
    __syncthreads();
    if (t == 0) {
        float tt = 0.0f; int dd = 0;
        for (int i = 0; i < C_ * B_; ++i) { tt += stotal[i]; dd += sdenom[i]; }
        out[0] = (dd > 0) ? (tt / (float)dd) : 0.0f;
    }
}

extern "C" void kernel_launch(void* const* d_in, const int* in_sizes, int n_in,
                              void* d_out, int out_size, void* d_ws, size_t ws_size,
                              hipStream_t stream) {
    const float* act    = (const float*)d_in[0];   // [P,B,N,G] f32
    const int*   labels = (const int*)  d_in[1];   // [B,H,W] i32
    const float* proto  = (const float*)d_in[2];   // [120,12] f32
    const float* gci    = (const float*)d_in[3];   // [16,12] f32
    float* out = (float*)d_out;
    float* ws  = (float*)d_ws;

    lse_count_kernel<<<C_ * B_, 256, 0, stream>>>(act, labels, gci, ws);
    t_wmma_kernel   <<<C_ * B_, 256, 0, stream>>>(act, labels, gci, ws);
    finalize_kernel <<<1, C_ * B_, 0, stream>>>(proto, ws, out);
}